// Decoder_51874615001600
// MI455X (gfx1250) — compile-verified
//
#include <hip/hip_runtime.h>
#include <hip/hip_bf16.h>
#include <cstdint>

typedef __attribute__((ext_vector_type(2))) float v2f;
typedef __attribute__((ext_vector_type(8))) float v8f;

#define B_SZ   128
#define S_LEN  400
#define ENC_D  1024
#define DEC_D  300
#define EMB_D  128
#define G3     900      // 3*DEC
#define RIN_D  1152     // EMB+ENC
#define HW_D   1324     // DEC+ENC
#define VOCAB_N 50000
#define PTOT   50050    // VOCAB + MAX_OOV

#define KCF    64       // K-chunk in floats
#define KC4    16       // K-chunk in float4 (k-steps per chunk)
#define LSTR   68       // LDS row stride in floats (64 + 4 pad -> bank step 4)

// ---------------------------------------------------------------------------
// Async global -> LDS copy (ASYNCcnt path) + wait, per CDNA5 ISA 15.18.3.
// LDS address = low 32 bits of the flat pointer (wave-relative LDS offset).
// ---------------------------------------------------------------------------
__device__ __forceinline__ void async_copy_b128(void* lds_dst, const void* gsrc)
{
    uint32_t lo = (uint32_t)(uintptr_t)lds_dst;
    asm volatile("global_load_async_to_lds_b128 %0, %1, off"
                 :: "v"(lo), "v"(gsrc) : "memory");
}
__device__ __forceinline__ void async_wait_all()
{
    asm volatile("s_wait_asynccnt 0x0" ::: "memory");
}

// ---------------------------------------------------------------------------
// C[m,n] = sum_k A[m,k]*B[n,k] (+ bias[n]).  M fixed at 128 (8 WMMA m-tiles).
// Block = 256 threads = 8 waves; wave w owns n-tile [n0, n0+16).
// Double-buffered LDS pipeline: chunk c+1 streams in via async-to-LDS while
// chunk c is consumed as ds_load_b64 + v_wmma_f32_16x16x4_f32.
// ---------------------------------------------------------------------------
__global__ __launch_bounds__(256) void gemm_at_bt_wmma(
    const float* __restrict__ A, const float* __restrict__ Bm,
    const float* __restrict__ bias, float* __restrict__ C, int N, int K)
{
    __shared__ float Ash[2][128 * LSTR];      // 2 x 34816 B
    __shared__ float Bsh[2][8][16 * LSTR];    // 2 x 34816 B

    const int tid  = threadIdx.x;
    const int wave = tid >> 5;
    const int lane = tid & 31;
    const int n0   = blockIdx.x * 128 + wave * 16;
    const bool wactive = (n0 < N);            // inactive waves still stage A + barrier

    const int  nl     = lane & 15;            // n within tile / m within A frag
    const int  kg     = lane >> 4;            // k-group 0/1
    const bool nvalid = wactive && (n0 + nl < N);
    const float nscale = nvalid ? 1.0f : 0.0f;

    const int K4 = K >> 2;                    // K % 4 == 0 for all call sites
    const float4* __restrict__ A4 = (const float4*)A;
    const float4* __restrict__ B4 = (const float4*)Bm;

    // ---- staging helpers -------------------------------------------------
    auto stageA = [&](int buf, int kc, int cl4) {        // cooperative, 256 thr
        const int kc4 = kc >> 2;
        for (int i = tid; i < 128 * cl4; i += 256) {
            int r, c;
            if (cl4 == KC4) { r = i >> 4; c = i & 15; }
            else            { r = i / cl4; c = i - r * cl4; }
            async_copy_b128(&Ash[buf][r * LSTR + c * 4],
                            A4 + ((size_t)r * K4 + kc4 + c));
        }
    };
    auto stageB = [&](int buf, int kc, int cl4) {        // per-wave private
        if (!wactive) return;
        const int kc4 = kc >> 2;
        for (int i = lane; i < 16 * cl4; i += 32) {
            int r, c;
            if (cl4 == KC4) { r = i >> 4; c = i & 15; }
            else            { r = i / cl4; c = i - r * cl4; }
            int grow = n0 + r; if (grow > N - 1) grow = N - 1;   // clamp tail
            async_copy_b128(&Bsh[buf][wave][r * LSTR + c * 4],
                            B4 + ((size_t)grow * K4 + kc4 + c));
        }
    };

    v8f acc[8] = {};

    auto compute = [&](int buf, int cl4) {
        if (!wactive) return;
        const float* __restrict__ Bp  = &Bsh[buf][wave][nl * LSTR + kg * 2];
        const float* __restrict__ Apb = &Ash[buf][nl * LSTR + kg * 2];
        #pragma unroll 4
        for (int ks = 0; ks < cl4; ++ks) {
            const int kof = ks * 4;
            v2f bf;
            bf.x = Bp[kof]     * nscale;
            bf.y = Bp[kof + 1] * nscale;
            #pragma unroll
            for (int mt = 0; mt < 8; ++mt) {
                const float* ap = Apb + mt * 16 * LSTR + kof;
                v2f af; af.x = ap[0]; af.y = ap[1];
                acc[mt] = __builtin_amdgcn_wmma_f32_16x16x4_f32(
                    false, af, false, bf, (short)0, acc[mt], false, false);
            }
        }
    };

    // ---- pipeline --------------------------------------------------------
    const int nch = (K + KCF - 1) / KCF;
    int cl4_0 = K < KCF ? (K >> 2) : KC4;
    stageA(0, 0, cl4_0);
    stageB(0, 0, cl4_0);
    async_wait_all();
    __syncthreads();

    for (int c = 0; c < nch; ++c) {
        const int kc  = c * KCF;
        const int rem = K - kc;
        const int cl4 = rem < KCF ? (rem >> 2) : KC4;
        if (c + 1 < nch) {
            const int kn   = kc + KCF;
            const int remn = K - kn;
            const int cl4n = remn < KCF ? (remn >> 2) : KC4;
            stageA((c + 1) & 1, kn, cl4n);
            stageB((c + 1) & 1, kn, cl4n);
        }
        compute(c & 1, cl4);
        async_wait_all();
        __syncthreads();
    }

    // ---- store -----------------------------------------------------------
    if (!nvalid) return;
    const int n = n0 + nl;
    const float bv = bias ? bias[n] : 0.0f;
    #pragma unroll
    for (int mt = 0; mt < 8; ++mt) {
        const int mb = mt * 16 + kg * 8;
        #pragma unroll
        for (int r = 0; r < 8; ++r)
            C[(size_t)(mb + r) * N + n] = acc[mt][r] + bv;
    }
}

// ---------------------------------------------------------------------------
// Fused attention: one block per batch row b.
// ---------------------------------------------------------------------------
__global__ __launch_bounds__(256) void attn_fused(
    const float* __restrict__ v, const float* __restrict__ enc,
    const unsigned char* __restrict__ mask,
    const float* __restrict__ copy_src, int copy_len,
    float* __restrict__ out, int col_off, int out_stride,
    float* __restrict__ a_out)
{
    const int b    = blockIdx.x;
    const int tid  = threadIdx.x;
    const int wave = tid >> 5;
    const int lane = tid & 31;

    __shared__ float vsh[ENC_D];
    __shared__ float sc[S_LEN];
    __shared__ float red[256];

    for (int i = tid; i < ENC_D; i += 256) vsh[i] = v[(size_t)b * ENC_D + i];
    __syncthreads();

    const float4* __restrict__ vsh4 = (const float4*)vsh;
    const float4* __restrict__ encb = (const float4*)(enc + (size_t)b * S_LEN * ENC_D);

    // pass 1: scores
    for (int s = wave; s < S_LEN; s += 8) {
        const float4* er = encb + (size_t)s * (ENC_D / 4);
        float acc = 0.0f;
        #pragma unroll
        for (int j = 0; j < ENC_D / 128; ++j) {
            const int idx = j * 32 + lane;
            float4 e4 = er[idx];
            float4 v4 = vsh4[idx];
            acc += e4.x * v4.x + e4.y * v4.y + e4.z * v4.z + e4.w * v4.w;
        }
        #pragma unroll
        for (int off = 16; off > 0; off >>= 1) acc += __shfl_xor(acc, off, 32);
        if (lane == 0)
            sc[s] = mask[(size_t)b * S_LEN + s] ? -__builtin_inff() : acc;
    }
    __syncthreads();

    // softmax over S
    float lm = -__builtin_inff();
    for (int s = tid; s < S_LEN; s += 256) lm = fmaxf(lm, sc[s]);
    red[tid] = lm; __syncthreads();
    for (int off = 128; off > 0; off >>= 1) {
        if (tid < off) red[tid] = fmaxf(red[tid], red[tid + off]);
        __syncthreads();
    }
    const float mx = red[0]; __syncthreads();

    float ls = 0.0f;
    for (int s = tid; s < S_LEN; s += 256) {
        float e = __expf(sc[s] - mx);
        sc[s] = e; ls += e;
    }
    red[tid] = ls; __syncthreads();
    for (int off = 128; off > 0; off >>= 1) {
        if (tid < off) red[tid] += red[tid + off];
        __syncthreads();
    }
    const float inv = 1.0f / red[0];
    for (int s = tid; s < S_LEN; s += 256) {
        float a = sc[s] * inv;
        sc[s] = a;
        if (a_out) a_out[(size_t)b * S_LEN + s] = a;
    }
    __syncthreads();

    // pass 2: weighted sum
    float4 w = {0.0f, 0.0f, 0.0f, 0.0f};
    for (int s = 0; s < S_LEN; ++s) {
        const float a = sc[s];
        float4 e4 = encb[(size_t)s * (ENC_D / 4) + tid];
        w.x += a * e4.x; w.y += a * e4.y; w.z += a * e4.z; w.w += a * e4.w;
    }
    float* orow = out + (size_t)b * out_stride + col_off;
    ((float4*)orow)[tid] = w;

    if (copy_src) {
        for (int i = tid; i < copy_len; i += 256)
            out[(size_t)b * out_stride + i] = copy_src[(size_t)b * copy_len + i];
    }
}

// ---------------------------------------------------------------------------
// GRU gate nonlinearity: h_new = (1-z)*n + z*h1 ; also writes hw[:,0:DEC].
// ---------------------------------------------------------------------------
__global__ __launch_bounds__(256) void gru_elem(
    const float* __restrict__ gx, const float* __restrict__ gh,
    const float* __restrict__ h1, float* __restrict__ h_new,
    float* __restrict__ hw)
{
    const int idx = blockIdx.x * blockDim.x + threadIdx.x;
    if (idx >= B_SZ * DEC_D) return;
    const int b = idx / DEC_D;
    const int j = idx - b * DEC_D;
    const float* gxr = gx + (size_t)b * G3;
    const float* ghr = gh + (size_t)b * G3;
    const float r = 1.0f / (1.0f + __expf(-(gxr[j] + ghr[j])));
    const float z = 1.0f / (1.0f + __expf(-(gxr[DEC_D + j] + ghr[DEC_D + j])));
    const float n = tanhf(gxr[2 * DEC_D + j] + r * ghr[2 * DEC_D + j]);
    const float h = (1.0f - z) * n + z * h1[idx];
    h_new[idx] = h;
    hw[(size_t)b * HW_D + j] = h;
}

// ---------------------------------------------------------------------------
// Per-row: p_gen, vocab softmax, assemble p_out row, pointer scatter.
// ---------------------------------------------------------------------------
__global__ __launch_bounds__(256) void output_assemble(
    const float* __restrict__ logits, const float* __restrict__ hw,
    const float* __restrict__ W_p, const float* __restrict__ b_p,
    const float* __restrict__ a2, const int* __restrict__ src_idx,
    float* __restrict__ p_out)
{
    const int b = blockIdx.x;
    const int tid = threadIdx.x;
    __shared__ float red[256];
    __shared__ float sh_pg;

    // p_gen
    const float* hwr = hw + (size_t)b * HW_D;
    float acc = 0.0f;
    for (int i = tid; i < HW_D; i += 256) acc += hwr[i] * W_p[i];
    red[tid] = acc; __syncthreads();
    for (int off = 128; off > 0; off >>= 1) {
        if (tid < off) red[tid] += red[tid + off];
        __syncthreads();
    }
    if (tid == 0) sh_pg = 1.0f / (1.0f + __expf(-(red[0] + b_p[0])));
    __syncthreads();
    const float pg = sh_pg;
    __syncthreads();

    // vocab softmax
    const float* lr = logits + (size_t)b * VOCAB_N;
    float lm = -__builtin_inff();
    for (int i = tid; i < VOCAB_N; i += 256) lm = fmaxf(lm, lr[i]);
    red[tid] = lm; __syncthreads();
    for (int off = 128; off > 0; off >>= 1) {
        if (tid < off) red[tid] = fmaxf(red[tid], red[tid + off]);
        __syncthreads();
    }
    const float mx = red[0]; __syncthreads();

    float ls = 0.0f;
    for (int i = tid; i < VOCAB_N; i += 256) ls += __expf(lr[i] - mx);
    red[tid] = ls; __syncthreads();
    for (int off = 128; off > 0; off >>= 1) {
        if (tid < off) red[tid] += red[tid + off];
        __syncthreads();
    }
    const float inv = 1.0f / red[0];

    float* por = p_out + (size_t)b * PTOT;
    for (int i = tid; i < VOCAB_N; i += 256)
        por[i] = pg * __expf(lr[i] - mx) * inv;
    for (int i = VOCAB_N + tid; i < PTOT; i += 256)
        por[i] = 0.0f;
    __threadfence_block();
    __syncthreads();

    const float omp = 1.0f - pg;
    for (int s = tid; s < S_LEN; s += 256)
        atomicAdd(&por[src_idx[(size_t)b * S_LEN + s]],
                  omp * a2[(size_t)b * S_LEN + s]);
}

// ---------------------------------------------------------------------------
extern "C" void kernel_launch(void* const* d_in, const int* in_sizes, int n_in,
                              void* d_out, int out_size, void* d_ws, size_t ws_size,
                              hipStream_t stream)
{
    (void)in_sizes; (void)n_in; (void)out_size; (void)ws_size;

    const float*         embedded = (const float*)d_in[0];
    const float*         enc      = (const float*)d_in[1];
    const float*         h1       = (const float*)d_in[2];
    const unsigned char* mask     = (const unsigned char*)d_in[3];
    /* d_in[4] = max_n_oov scalar (hardcoded 50) */
    const int*           src_idx  = (const int*)d_in[5];
    const float*         W_attn1  = (const float*)d_in[6];
    const float*         W_attn2  = (const float*)d_in[7];
    const float*         W_ih     = (const float*)d_in[8];
    const float*         W_hh     = (const float*)d_in[9];
    const float*         b_ih     = (const float*)d_in[10];
    const float*         b_hh     = (const float*)d_in[11];
    const float*         W_out    = (const float*)d_in[12];
    const float*         b_out    = (const float*)d_in[13];
    const float*         W_p      = (const float*)d_in[14];
    const float*         b_p      = (const float*)d_in[15];

    float* ws      = (float*)d_ws;
    float* v1      = ws;                          // 128*1024
    float* rnn_in  = v1     + B_SZ * ENC_D;       // 128*1152
    float* gx      = rnn_in + B_SZ * RIN_D;       // 128*900
    float* gh      = gx     + B_SZ * G3;          // 128*900
    float* v2      = gh     + B_SZ * G3;          // 128*1024
    float* hw      = v2     + B_SZ * ENC_D;       // 128*1324
    float* a2      = hw     + B_SZ * HW_D;        // 128*400
    float* logits  = a2     + B_SZ * S_LEN;       // 128*50000

    float* p_out = (float*)d_out;
    float* h_new = p_out + (size_t)B_SZ * PTOT;   // second output, [128,300]

    const dim3 blk(256);
    auto nblocks = [](int N) { return dim3((unsigned)((N + 127) / 128)); };

    // attn1: v1 = h1 @ W_attn1^T
    gemm_at_bt_wmma<<<nblocks(ENC_D), blk, 0, stream>>>(h1, W_attn1, nullptr, v1, ENC_D, DEC_D);
    // attn1 fused -> rnn_input = [embedded | weighted]
    attn_fused<<<dim3(B_SZ), blk, 0, stream>>>(v1, enc, mask, embedded, EMB_D,
                                               rnn_in, EMB_D, RIN_D, nullptr);
    // GRU gates
    gemm_at_bt_wmma<<<nblocks(G3), blk, 0, stream>>>(rnn_in, W_ih, b_ih, gx, G3, RIN_D);
    gemm_at_bt_wmma<<<nblocks(G3), blk, 0, stream>>>(h1, W_hh, b_hh, gh, G3, DEC_D);
    gru_elem<<<dim3((B_SZ * DEC_D + 255) / 256), blk, 0, stream>>>(gx, gh, h1, h_new, hw);
    // attn2: v2 = h_new @ W_attn2^T ; fused -> hw[:,300:] and a2
    gemm_at_bt_wmma<<<nblocks(ENC_D), blk, 0, stream>>>(h_new, W_attn2, nullptr, v2, ENC_D, DEC_D);
    attn_fused<<<dim3(B_SZ), blk, 0, stream>>>(v2, enc, mask, nullptr, 0,
                                               hw, DEC_D, HW_D, a2);
    // output projection (dominant GEMM, 265 MB of W_out streamed)
    gemm_at_bt_wmma<<<nblocks(VOCAB_N), blk, 0, stream>>>(hw, W_out, b_out, logits, VOCAB_N, HW_D);
    // softmax + p_gen + pointer scatter
    output_assemble<<<dim3(B_SZ), blk, 0, stream>>>(logits, hw, W_p, b_p, a2, src_idx, p_out);
}